// Model_17145509445961
// MI455X (gfx1250) — compile-verified
//
#include <hip/hip_runtime.h>

// HMM scaled-forward on gfx1250 (MI455X), log-semiring mapped to f32 WMMA.
//
//   alpha_t = alpha_{t-1} @ (A * diag(E[:, o_t]))          (prob space)
//   loglik  = sum_t log c_t  (scale factors; logs telescope exactly)
//
// Two-pass chunked scan (needs ~16.9 MB scratch, gated on ws_size):
//   pass 1: per (batch row, 128-step chunk): scaled 16x16 matrix product
//           P_c = prod_t A*diag(e_t) via chains of V_WMMA_F32_16X16X4_F32.
//           16384 independent waves -> fills the chip. Normalization (cross-
//           lane sum + rcp + log2) deferred to every 4th step: f32 range
//           analysis bounds 4 unnormalized steps to >= ~2^-60 (safe).
//   pass 2: per batch row: alpha = alpha @ P_c over 32 chunks (shuffle-FMA).
// Fallback (small ws): single-pass sequential scan, one wave per 16-batch
// tile, same 4-step deferred normalization.

typedef float v2f __attribute__((ext_vector_type(2)));
typedef float v8f __attribute__((ext_vector_type(8)));

namespace {
constexpr int kS = 16;         // hidden states
constexpr int kO = 64;         // observation symbols
constexpr int kB = 512;        // batch
constexpr int kT = 4096;       // time steps
constexpr int kC = 128;        // chunk length (pass-1 task depth)
constexpr int kNC = kT / kC;   // 32 chunks per row
constexpr int kEStride = 20;   // padded LDS row stride (floats), conflict-free
constexpr int kTStride = 20;
// ws layout (floats): [0..15] pi | [16..271] A | [272..1295] E[o][s] |
//                     [kPOff ..) chunk matrices | [kLgsOff ..) chunk log2-scales
constexpr int kPOff   = 2048;
constexpr int kLgsOff = kPOff + kB * kNC * kS * kS;          // 2048 + 4194304
constexpr size_t kWsNeed = (size_t)(kLgsOff + kB * kNC) * 4; // ~16.9 MB
constexpr float kLn2 = 0.69314718055994530942f;
}

// ---------------- pass 0: log-params -> probability tables -----------------
__global__ __launch_bounds__(64)
void hmm_prep(const float* __restrict__ lpi,
              const float* __restrict__ ltr,
              const float* __restrict__ lem,
              float* __restrict__ ws) {
  const int tid = threadIdx.x;
  float* __restrict__ pi = ws;
  float* __restrict__ A  = ws + 16;
  float* __restrict__ E  = ws + 16 + kS * kS;

  if (tid < kS) {
    float mx = -INFINITY;
    for (int n = 0; n < kS; ++n) mx = fmaxf(mx, ltr[tid * kS + n]);
    float s = 0.f;
    for (int n = 0; n < kS; ++n) s += expf(ltr[tid * kS + n] - mx);
    const float inv = 1.f / s;
    for (int n = 0; n < kS; ++n)
      A[tid * kS + n] = expf(ltr[tid * kS + n] - mx) * inv;

    float mxe = -INFINITY;
    for (int o = 0; o < kO; ++o) mxe = fmaxf(mxe, lem[tid * kO + o]);
    float se = 0.f;
    for (int o = 0; o < kO; ++o) se += expf(lem[tid * kO + o] - mxe);
    const float inve = 1.f / se;
    for (int o = 0; o < kO; ++o)
      E[o * kS + tid] = expf(lem[tid * kO + o] - mxe) * inve;   // transposed
  } else if (tid == kS) {
    float mx = -INFINITY;
    for (int n = 0; n < kS; ++n) mx = fmaxf(mx, lpi[n]);
    float s = 0.f;
    for (int n = 0; n < kS; ++n) s += expf(lpi[n] - mx);
    const float inv = 1.f / s;
    for (int n = 0; n < kS; ++n) pi[n] = expf(lpi[n] - mx) * inv;
  }
}

// ---------------- pass 1: per-(row,chunk) scaled matrix product -------------
// 8 waves / block share the LDS emission table; each wave owns one task and a
// private transpose tile (wave-local LDS ordering => no block barrier needed).
__global__ __launch_bounds__(256)
void hmm_chunks(const int* __restrict__ obs, float* __restrict__ ws) {
  __shared__ float eL[kO * kEStride];
  __shared__ float tiles[8 * kS * kTStride];

  const int tid  = threadIdx.x;
  const int lane = tid & 31;
  const int wid  = tid >> 5;
  const int n16  = lane & 15;
  const int hh   = lane >> 4;
  const int off  = hh * 2;

  const float* __restrict__ AG = ws + 16;
  const float* __restrict__ EG = ws + 16 + kS * kS;

  for (int idx = tid; idx < kO * kS; idx += 256)
    eL[(idx >> 4) * kEStride + (idx & 15)] = EG[idx];
  __syncthreads();

  float* __restrict__ tileW = tiles + wid * (kS * kTStride);

  const int task = blockIdx.x * 8 + wid;   // 0 .. kB*kNC-1
  const int row  = task >> 5;              // batch row
  const int c    = task & (kNC - 1);       // chunk index
  const int* __restrict__ obsC = obs + row * kT + c * kC;

  // Static B operand: slice k, elem j on lane L = A[4k + 2*hh + j][n16]
  v2f b0, b1, b2, b3;
  b0.x = AG[(off +  0) * kS + n16]; b0.y = AG[(off +  1) * kS + n16];
  b1.x = AG[(off +  4) * kS + n16]; b1.y = AG[(off +  5) * kS + n16];
  b2.x = AG[(off +  8) * kS + n16]; b2.y = AG[(off +  9) * kS + n16];
  b3.x = AG[(off + 12) * kS + n16]; b3.y = AG[(off + 13) * kS + n16];

  v2f a0, a1, a2, a3;      // P in WMMA A-operand layout (lane owns row n16)
  float w[kS];             // full P row (only valid on normalizing steps)
  float rs  = 1.f;         // 1/total at last normalization
  float lc2 = 0.f;         // accumulated log2 of scales

  auto reduceScale = [&]() -> float {     // sum all 256 entries; set rs
    const float t0 = (w[ 0] + w[ 1]) + (w[ 2] + w[ 3]);
    const float t1 = (w[ 4] + w[ 5]) + (w[ 6] + w[ 7]);
    const float t2 = (w[ 8] + w[ 9]) + (w[10] + w[11]);
    const float t3 = (w[12] + w[13]) + (w[14] + w[15]);
    float tot = (t0 + t1) + (t2 + t3);    // row sum (depth-4 tree)
    tot += __shfl_xor(tot, 1, 32);        // + across the 16 rows
    tot += __shfl_xor(tot, 2, 32);
    tot += __shfl_xor(tot, 4, 32);
    tot += __shfl_xor(tot, 8, 32);
    rs = __builtin_amdgcn_rcpf(tot);
    return tot;
  };
  auto foldScaled = [&]() {               // scaled row -> A-operand slices
    a0.x = (hh ? w[ 2] : w[ 0]) * rs; a0.y = (hh ? w[ 3] : w[ 1]) * rs;
    a1.x = (hh ? w[ 6] : w[ 4]) * rs; a1.y = (hh ? w[ 7] : w[ 5]) * rs;
    a2.x = (hh ? w[10] : w[ 8]) * rs; a2.y = (hh ? w[11] : w[ 9]) * rs;
    a3.x = (hh ? w[14] : w[12]) * rs; a3.y = (hh ? w[15] : w[13]) * rs;
  };

  // ---- init: P = diag(e_{t0}) for chunk 0, else A * diag(e_{t0}) ----
  const int4 g0 = *(const int4*)(obsC);
  if (c == 0) {
    const float ev = eL[g0.x * kEStride + n16];
#pragma unroll
    for (int n = 0; n < kS; ++n) w[n] = (n == n16) ? ev : 0.f;
  } else {
    const float* __restrict__ ep = eL + g0.x * kEStride;
#pragma unroll
    for (int n = 0; n < kS; ++n) w[n] = AG[n16 * kS + n] * ep[n];
  }
  lc2 = __log2f(reduceScale());
  foldScaled();

  // ---- one product step: P <- (P @ A) * diag(e_t) -----------------------
  // norm=true: renormalize (sum -> rcp -> log2). norm=false: fast path,
  // re-fold the 8 needed columns straight from the LDS tile (4x ds_load_b64).
  auto step = [&](int o, bool norm) {
    v8f acc = {0.f, 0.f, 0.f, 0.f, 0.f, 0.f, 0.f, 0.f};
    acc = __builtin_amdgcn_wmma_f32_16x16x4_f32(false, a0, false, b0, (short)0, acc, false, false);
    acc = __builtin_amdgcn_wmma_f32_16x16x4_f32(false, a1, false, b1, (short)0, acc, false, false);
    acc = __builtin_amdgcn_wmma_f32_16x16x4_f32(false, a2, false, b2, (short)0, acc, false, false);
    acc = __builtin_amdgcn_wmma_f32_16x16x4_f32(false, a3, false, b3, (short)0, acc, false, false);
    // D layout: (m = r+8*hh, n = n16); emission is a column factor => one load
    const float ev = eL[o * kEStride + n16];
#pragma unroll
    for (int r = 0; r < 8; ++r)
      tileW[(r + 8 * hh) * kTStride + n16] = acc[r] * ev;
    asm volatile("s_wait_dscnt 0x0" ::: "memory");  // wave-local LDS fence
    if (norm) {
      const float* __restrict__ rowp = tileW + n16 * kTStride;
#pragma unroll
      for (int n = 0; n < kS; ++n) w[n] = rowp[n];
      lc2 += __log2f(reduceScale());
      foldScaled();
    } else {
      const float* __restrict__ rowp = tileW + n16 * kTStride + off;
      a0.x = rowp[ 0]; a0.y = rowp[ 1];
      a1.x = rowp[ 4]; a1.y = rowp[ 5];
      a2.x = rowp[ 8]; a2.y = rowp[ 9];
      a3.x = rowp[12]; a3.y = rowp[13];
    }
  };

  step(g0.y, false); step(g0.z, false); step(g0.w, true);   // 3 steps
#pragma unroll 1
  for (int tb = 4; tb < kC; tb += 4) {                      // + 31*4 = 127
    const int4 g = *(const int4*)(obsC + tb);
    step(g.x, false); step(g.y, false); step(g.z, false); step(g.w, true);
  }

  // ---- emit normalized chunk matrix (row-major) + its log2 scale ----
  float* __restrict__ Pg = ws + kPOff + (size_t)(row * kNC + c) * (kS * kS);
  if (hh == 0) {
#pragma unroll
    for (int n = 0; n < kS; ++n) Pg[n16 * kS + n] = w[n] * rs;
  }
  if (lane == 0) ws[kLgsOff + row * kNC + c] = lc2;
}

// ---------------- pass 2: per-row combine over 32 chunks --------------------
// Half-wave per batch row: lane owns state n; alpha @ P via broadcast-FMA,
// double-buffered chunk-matrix loads.
__global__ __launch_bounds__(32)
void hmm_combine(const float* __restrict__ ws, float* __restrict__ out) {
  const int lane = threadIdx.x;
  const int n    = lane & 15;
  const int row  = blockIdx.x * 2 + (lane >> 4);

  const float* __restrict__ Pb  = ws + kPOff + (size_t)row * kNC * (kS * kS);
  const float* __restrict__ lgs = ws + kLgsOff + row * kNC;

  float alpha = ws[n];   // pi (normalized; chunk 0 already holds diag(e_0))
  float lc2 = 0.f;

  float p[kS];
#pragma unroll
  for (int i = 0; i < kS; ++i) p[i] = Pb[i * kS + n];

#pragma unroll 1
  for (int c = 0; c < kNC; ++c) {
    const int cn = (c + 1 < kNC) ? (c + 1) : c;     // branch-free prefetch
    const float* __restrict__ Pn = Pb + cn * (kS * kS);
    float pn[kS];
#pragma unroll
    for (int i = 0; i < kS; ++i) pn[i] = Pn[i * kS + n];

    float s = 0.f;
#pragma unroll
    for (int i = 0; i < kS; ++i)
      s = fmaf(__shfl(alpha, (lane & 16) + i, 32), p[i], s);

    float tot = s;
    tot += __shfl_xor(tot, 1, 32);
    tot += __shfl_xor(tot, 2, 32);
    tot += __shfl_xor(tot, 4, 32);
    tot += __shfl_xor(tot, 8, 32);
    alpha = s * __builtin_amdgcn_rcpf(tot);
    lc2 += __log2f(tot) + lgs[c];

#pragma unroll
    for (int i = 0; i < kS; ++i) p[i] = pn[i];
  }
  // alpha sums to 1 each step => final logsumexp == accumulated log scales
  if (n == 0) out[row] = lc2 * kLn2;
}

// ---------------- fallback: single-pass scan (small scratch) ----------------
__global__ __launch_bounds__(32)
void hmm_forward(const int* __restrict__ obs,
                 const float* __restrict__ ws,
                 float* __restrict__ out) {
  __shared__ float eL[kO * kEStride];
  __shared__ float tile[kS * kTStride];

  const int lane = threadIdx.x;
  const int n16  = lane & 15;
  const int hh   = lane >> 4;
  const int off  = hh * 2;

  const float* __restrict__ piG = ws;
  const float* __restrict__ AG  = ws + 16;
  const float* __restrict__ EG  = ws + 16 + kS * kS;

  for (int idx = lane; idx < kO * kS; idx += 32)
    eL[(idx >> 4) * kEStride + (idx & 15)] = EG[idx];
  __syncthreads();

  v2f b0, b1, b2, b3;
  b0.x = AG[(off +  0) * kS + n16]; b0.y = AG[(off +  1) * kS + n16];
  b1.x = AG[(off +  4) * kS + n16]; b1.y = AG[(off +  5) * kS + n16];
  b2.x = AG[(off +  8) * kS + n16]; b2.y = AG[(off +  9) * kS + n16];
  b3.x = AG[(off + 12) * kS + n16]; b3.y = AG[(off + 13) * kS + n16];

  const int mrow = blockIdx.x * kS + n16;
  const int* __restrict__ obsRow = obs + mrow * kT;

  v2f a0, a1, a2, a3;
  float lc2 = 0.f, kc = 0.f;   // Kahan, log2 domain
  float w[kS];
  float rs = 1.f;

  auto reduceRow = [&]() -> float {       // per-row sum (diagonal scale is OK)
    const float t0 = (w[ 0] + w[ 1]) + (w[ 2] + w[ 3]);
    const float t1 = (w[ 4] + w[ 5]) + (w[ 6] + w[ 7]);
    const float t2 = (w[ 8] + w[ 9]) + (w[10] + w[11]);
    const float t3 = (w[12] + w[13]) + (w[14] + w[15]);
    const float s = (t0 + t1) + (t2 + t3);
    rs = __builtin_amdgcn_rcpf(s);
    return s;
  };
  auto foldScaled = [&]() {
    a0.x = (hh ? w[ 2] : w[ 0]) * rs; a0.y = (hh ? w[ 3] : w[ 1]) * rs;
    a1.x = (hh ? w[ 6] : w[ 4]) * rs; a1.y = (hh ? w[ 7] : w[ 5]) * rs;
    a2.x = (hh ? w[10] : w[ 8]) * rs; a2.y = (hh ? w[11] : w[ 9]) * rs;
    a3.x = (hh ? w[14] : w[12]) * rs; a3.y = (hh ? w[15] : w[13]) * rs;
  };

  const int4 g0 = *(const int4*)(obsRow);
  {
    const float* __restrict__ ep = eL + g0.x * kEStride;
#pragma unroll
    for (int nn = 0; nn < kS; ++nn) w[nn] = piG[nn] * ep[nn];
    lc2 = __log2f(reduceRow());
    foldScaled();
  }

  auto step = [&](int o, bool norm) {
    v8f acc = {0.f, 0.f, 0.f, 0.f, 0.f, 0.f, 0.f, 0.f};
    acc = __builtin_amdgcn_wmma_f32_16x16x4_f32(false, a0, false, b0, (short)0, acc, false, false);
    acc = __builtin_amdgcn_wmma_f32_16x16x4_f32(false, a1, false, b1, (short)0, acc, false, false);
    acc = __builtin_amdgcn_wmma_f32_16x16x4_f32(false, a2, false, b2, (short)0, acc, false, false);
    acc = __builtin_amdgcn_wmma_f32_16x16x4_f32(false, a3, false, b3, (short)0, acc, false, false);
#pragma unroll
    for (int r = 0; r < 8; ++r)
      tile[(r + 8 * hh) * kTStride + n16] = acc[r];
    __syncthreads();
    if (norm) {
      const float* __restrict__ rowp = tile + n16 * kTStride;
      const float* __restrict__ ep   = eL + o * kEStride;
#pragma unroll
      for (int nn = 0; nn < kS; ++nn) w[nn] = rowp[nn] * ep[nn];
      const float lg = __log2f(reduceRow());
      const float y  = lg - kc;       // Kahan (one log per 4-step group)
      const float t2 = lc2 + y;
      kc = (t2 - lc2) - y;
      lc2 = t2;
      foldScaled();
    } else {
      const float* __restrict__ rowp = tile + n16 * kTStride + off;
      const float* __restrict__ ep   = eL + o * kEStride + off;
      a0.x = rowp[ 0] * ep[ 0]; a0.y = rowp[ 1] * ep[ 1];
      a1.x = rowp[ 4] * ep[ 4]; a1.y = rowp[ 5] * ep[ 5];
      a2.x = rowp[ 8] * ep[ 8]; a2.y = rowp[ 9] * ep[ 9];
      a3.x = rowp[12] * ep[12]; a3.y = rowp[13] * ep[13];
    }
  };

  int4 cur = *(const int4*)(obsRow + 4);
  step(g0.y, false); step(g0.z, false); step(g0.w, true);
#pragma unroll 1
  for (int tb = 4; tb < kT - 4; tb += 4) {
    const int4 nxt = *(const int4*)(obsRow + tb + 4);
    step(cur.x, false); step(cur.y, false); step(cur.z, false); step(cur.w, true);
    cur = nxt;
  }
  step(cur.x, false); step(cur.y, false); step(cur.z, false); step(cur.w, true);

  if (lane < 16) out[mrow] = lc2 * kLn2;
}

// ---------------------------------------------------------------------------
extern "C" void kernel_launch(void* const* d_in, const int* in_sizes, int n_in,
                              void* d_out, int out_size, void* d_ws, size_t ws_size,
                              hipStream_t stream) {
  (void)in_sizes; (void)n_in; (void)out_size;
  const int*   obs = (const int*)d_in[0];
  const float* lpi = (const float*)d_in[1];
  const float* ltr = (const float*)d_in[2];
  const float* lem = (const float*)d_in[3];
  float* out = (float*)d_out;
  float* ws  = (float*)d_ws;

  hmm_prep<<<1, 64, 0, stream>>>(lpi, ltr, lem, ws);
  if (ws_size >= kWsNeed) {
    // chip-filling path: 16384 WMMA waves, then tiny per-row combine
    hmm_chunks<<<(kB * kNC) / 8, 256, 0, stream>>>(obs, ws);
    hmm_combine<<<kB / 2, 32, 0, stream>>>(ws, out);
  } else {
    // scratch-lean path: one wave per 16-batch tile
    hmm_forward<<<kB / kS, 32, 0, stream>>>(obs, ws, out);
  }
}